// BiGramModel_71201967833514
// MI455X (gfx1250) — compile-verified
//
#include <hip/hip_runtime.h>
#include <math.h>

typedef float v2f __attribute__((ext_vector_type(2)));
typedef float v8f __attribute__((ext_vector_type(8)));

// Model dims (fixed by the reference)
constexpr int CB = 8;       // batch
constexpr int CT = 128;     // block size
constexpr int CE = 256;     // embedding (== num heads, head_size 1)
constexpr int CV = 8192;    // vocab
constexpr int CL = 5;       // layers
constexpr int CM = CB * CT; // 1024 rows

// -------------------------------------------------------------------------
// Embedding: x[b,t,:] = emb[idx[b,t],:] + pos_emb[t,:]
// -------------------------------------------------------------------------
__global__ void embed_k(const int* __restrict__ idxs,
                        const float* __restrict__ emb,
                        const float* __restrict__ pos,
                        float* __restrict__ x) {
  int i = blockIdx.x * blockDim.x + threadIdx.x;
  if (i >= CM * CE) return;
  int e  = i & (CE - 1);
  int bt = i >> 8;            // / CE
  int t  = bt & (CT - 1);
  int tok = idxs[bt];
  x[i] = emb[tok * CE + e] + pos[t * CE + e];
}

// -------------------------------------------------------------------------
// WMMA f32 GEMM: C[M,N] = A[M,K] @ B[K,N]  (row-major), one 16x16 tile/wave.
// Wave32 fragment layouts per CDNA5 ISA 7.12.2:
//   A 16x4 : lane m=l&15; VGPR j = A[m, kk + 2*(l>>4) + j]
//   B 4x16 : lane n=l&15; VGPR j = B[kk + 2*(l>>4) + j, n]
//   C 16x16: lane n=l&15; VGPR r = C[r + 8*(l>>4), n]
// OP 0: store acc. OP 1: out = resid + relu(acc + bias). OP 2: out = acc + bias.
// -------------------------------------------------------------------------
template <int OP>
__global__ void gemm16_k(const float* __restrict__ A,
                         const float* __restrict__ Bm,
                         float* __restrict__ C,
                         const float* __restrict__ bias,
                         const float* __restrict__ resid,
                         int K, int N) {
  int wave = blockIdx.x * (blockDim.x >> 5) + (threadIdx.x >> 5);
  int lane = threadIdx.x & 31;
  int tilesN = N >> 4;
  int tm = wave / tilesN;
  int tn = wave - tm * tilesN;
  int lo = lane & 15;
  int hi = lane >> 4;                 // 0 or 1
  int row = tm * 16 + lo;             // A row for this lane
  int col = tn * 16 + lo;             // B/C col for this lane

  v8f acc = {};
  const float* Arow = A + (size_t)row * K + hi * 2;
  for (int kk = 0; kk < K; kk += 4) {
    v2f a = *(const v2f*)(Arow + kk);
    v2f b;
    int kb = kk + hi * 2;
    b.x = Bm[(size_t)(kb + 0) * N + col];
    b.y = Bm[(size_t)(kb + 1) * N + col];
    acc = __builtin_amdgcn_wmma_f32_16x16x4_f32(
        /*neg_a=*/false, a, /*neg_b=*/false, b,
        /*c_mod=*/(short)0, acc, /*reuse_a=*/false, /*reuse_b=*/false);
  }

#pragma unroll
  for (int r = 0; r < 8; ++r) {
    int orow = tm * 16 + r + hi * 8;
    float val = acc[r];
    if (OP == 1) {
      float t = val + bias[col];
      t = t > 0.f ? t : 0.f;
      val = resid[(size_t)orow * N + col] + t;
    } else if (OP == 2) {
      val = val + bias[col];
    }
    C[(size_t)orow * N + col] = val;
  }
}

// -------------------------------------------------------------------------
// Attention for one (b,h): scores[i,j]=q_i*k_j, causal (j<=i), softmax over
// the QUERY axis i per column j (faithful quirk), out_i = sum_j a[i,j]*v_j.
// One 128-thread block per (b,h). Loop bounds apply the causal mask.
// -------------------------------------------------------------------------
__global__ void attn_k(const float* __restrict__ q,
                       const float* __restrict__ k,
                       const float* __restrict__ v,
                       float* __restrict__ out) {
  __shared__ float qs[CT], ks[CT], vs[CT], ms[CT], ss[CT];
  int bh = blockIdx.x;
  int h = bh & (CE - 1);
  int b = bh >> 8;
  int t = threadIdx.x;
  int base = b * CT * CE + h;
  qs[t] = q[base + t * CE];
  ks[t] = k[base + t * CE];
  vs[t] = v[base + t * CE];
  __syncthreads();

  // phase 1: column j = t; softmax stats over valid rows i in [j, T)
  float kj = ks[t];
  float m = -3.402823466e+38f;
  for (int i = t; i < CT; ++i) m = fmaxf(m, qs[i] * kj);
  float s = 0.f;
  for (int i = t; i < CT; ++i) s += expf(qs[i] * kj - m);
  ms[t] = m;
  ss[t] = s;
  __syncthreads();

  // phase 2: row i = t; out_i = sum_{j<=i} exp(q_i k_j - m_j)/s_j * v_j
  float qi = qs[t];
  float o = 0.f;
  for (int j = 0; j <= t; ++j) o += expf(qi * ks[j] - ms[j]) / ss[j] * vs[j];
  out[base + t * CE] = o;
}

// -------------------------------------------------------------------------
// Elementwise residual add: x += y
// -------------------------------------------------------------------------
__global__ void add_k(float* __restrict__ x, const float* __restrict__ y, int n) {
  int i = blockIdx.x * blockDim.x + threadIdx.x;
  if (i < n) x[i] += y[i];
}

// -------------------------------------------------------------------------
// Per-row NLL: rowloss[r] = logsumexp(logits[r,:]) - logits[r, target[r]]
// -------------------------------------------------------------------------
__global__ void loss_rows_k(const float* __restrict__ logits,
                            const int* __restrict__ targets,
                            float* __restrict__ rowloss) {
  __shared__ float red[256];
  int r = blockIdx.x;
  const float* lr = logits + (size_t)r * CV;
  int t = threadIdx.x;
  float m = -3.402823466e+38f;
  for (int j = t; j < CV; j += 256) m = fmaxf(m, lr[j]);
  red[t] = m;
  __syncthreads();
  for (int s = 128; s > 0; s >>= 1) {
    if (t < s) red[t] = fmaxf(red[t], red[t + s]);
    __syncthreads();
  }
  m = red[0];
  __syncthreads();
  float sum = 0.f;
  for (int j = t; j < CV; j += 256) sum += expf(lr[j] - m);
  red[t] = sum;
  __syncthreads();
  for (int s = 128; s > 0; s >>= 1) {
    if (t < s) red[t] += red[t + s];
    __syncthreads();
  }
  if (t == 0) {
    float lse = m + logf(red[0]);
    rowloss[r] = lse - lr[targets[r]];
  }
}

__global__ void loss_final_k(const float* __restrict__ rowloss,
                             float* __restrict__ out) {
  __shared__ float red[256];
  int t = threadIdx.x;
  float s = 0.f;
  for (int r = t; r < CM; r += 256) s += rowloss[r];
  red[t] = s;
  __syncthreads();
  for (int st = 128; st > 0; st >>= 1) {
    if (t < st) red[t] += red[t + st];
    __syncthreads();
  }
  if (t == 0) out[0] = red[0] / (float)CM;
}

// -------------------------------------------------------------------------
extern "C" void kernel_launch(void* const* d_in, const int* in_sizes, int n_in,
                              void* d_out, int out_size, void* d_ws, size_t ws_size,
                              hipStream_t stream) {
  (void)in_sizes; (void)n_in; (void)out_size; (void)ws_size;

  const int*   idxs    = (const int*)d_in[0];
  const int*   targets = (const int*)d_in[1];
  const float* emb     = (const float*)d_in[2];
  const float* pos     = (const float*)d_in[3];
  const float* Wq      = (const float*)d_in[4];
  const float* Wk      = (const float*)d_in[5];
  const float* Wv      = (const float*)d_in[6];
  const float* Wfc     = (const float*)d_in[7];
  const float* bfc     = (const float*)d_in[8];
  const float* Wlm     = (const float*)d_in[9];
  const float* blm     = (const float*)d_in[10];

  float* logits = (float*)d_out;
  float* loss   = logits + (size_t)CM * CV;

  float* ws = (float*)d_ws;
  float* xa = ws; ws += CM * CE;
  float* xb = ws; ws += CM * CE;
  float* q  = ws; ws += CM * CE;
  float* k  = ws; ws += CM * CE;
  float* v  = ws; ws += CM * CE;
  float* ao = ws; ws += CM * CE;
  float* rowloss = ws; ws += CM;

  embed_k<<<(CM * CE + 255) / 256, 256, 0, stream>>>(idxs, emb, pos, xa);

  const int tilesE  = (CM / 16) * (CE / 16);   // 1024 tiles -> 256 blocks
  const int tilesLM = (CM / 16) * (CV / 16);   // 32768 tiles -> 8192 blocks

  float* x  = xa;
  float* xn = xb;
  for (int l = 0; l < CL; ++l) {
    const float* wq = Wq + (size_t)l * CE * CE;
    const float* wk = Wk + (size_t)l * CE * CE;
    const float* wv = Wv + (size_t)l * CE * CE;
    const float* wf = Wfc + (size_t)l * CE * CE;
    const float* bf = bfc + (size_t)l * CE;

    gemm16_k<0><<<tilesE / 4, 128, 0, stream>>>(x, wq, q, nullptr, nullptr, CE, CE);
    gemm16_k<0><<<tilesE / 4, 128, 0, stream>>>(x, wk, k, nullptr, nullptr, CE, CE);
    gemm16_k<0><<<tilesE / 4, 128, 0, stream>>>(x, wv, v, nullptr, nullptr, CE, CE);

    attn_k<<<CB * CE, CT, 0, stream>>>(q, k, v, ao);
    add_k<<<(CM * CE + 255) / 256, 256, 0, stream>>>(x, ao, CM * CE);

    // xn = x + relu(x @ Wfc + bfc)
    gemm16_k<1><<<tilesE / 4, 128, 0, stream>>>(x, wf, xn, bf, x, CE, CE);
    float* tmp = x; x = xn; xn = tmp;
  }

  // logits = x @ Wlm + blm  (written straight to d_out)
  gemm16_k<2><<<tilesLM / 4, 128, 0, stream>>>(x, Wlm, logits, blm, nullptr, CE, CV);

  loss_rows_k<<<CM, 256, 0, stream>>>(logits, targets, rowloss);
  loss_final_k<<<1, 256, 0, stream>>>(rowloss, loss);
}